// GQA_20486994002327
// MI455X (gfx1250) — compile-verified
//
#include <hip/hip_runtime.h>
#include <math.h>
#include <stdint.h>

// ---------------------------------------------------------------------------
// Problem constants (match reference)
// ---------------------------------------------------------------------------
constexpr int Bc   = 2;
constexpr int Nn   = 2048;
constexpr int DIM  = 2048;
constexpr int Hh   = 32;    // N_HEAD
constexpr int KVH  = 8;     // N_KV_HEAD
constexpr int HD   = 64;    // HEAD_DIM
constexpr int ROWS = Bc * Nn;          // 4096 GEMM rows

typedef __attribute__((ext_vector_type(2))) float v2f;
typedef __attribute__((ext_vector_type(8))) float v8f;

// CDNA5 fp32 WMMA: D(16x16,f32) = A(16x4,f32) * B(4x16,f32) + C
static __device__ __forceinline__ v8f wmma4(v2f a, v2f b, v8f c) {
    return __builtin_amdgcn_wmma_f32_16x16x4_f32(
        /*neg_a=*/false, a, /*neg_b=*/false, b,
        /*c_mod=*/(short)0, c, /*reuse_a=*/false, /*reuse_b=*/false);
}

// gfx1250 async copy global->LDS (16B per lane), tracked by ASYNCcnt.
static __device__ __forceinline__ void async_b128(uint32_t lds_off, const float* g) {
    asm volatile("global_load_async_to_lds_b128 %0, %1, off"
                 :: "v"(lds_off), "v"(g) : "memory");
}

// ---------------------------------------------------------------------------
// Kernel 1: fp32 GEMM via V_WMMA_F32_16X16X4_F32, async-LDS double buffered.
// C[M x Nc] = A[M x K] * B[K x Nc], row-major. M%128==0, Nc%64==0, K%32==0.
// Block: 256 thr (8 waves). Block tile 128x64, wave tile 32x32, K-chunk 32.
// ---------------------------------------------------------------------------
#define KC    32
#define LDA_S 36   // A LDS row stride (floats): 16B-aligned chunks, conflict-free frags
#define LDB_S 72   // B LDS row stride (floats): half-groups (rows k, k+2) on disjoint banks

__global__ __launch_bounds__(256) void gemm_f32_wmma(
    const float* __restrict__ A, const float* __restrict__ B, float* __restrict__ C,
    int M, int Nc, int K)
{
    __shared__ float sA0[128 * LDA_S];
    __shared__ float sB0[KC  * LDB_S];
    __shared__ float sA1[128 * LDA_S];
    __shared__ float sB1[KC  * LDB_S];

    const int tid  = threadIdx.x;
    const int lane = tid & 31;
    const int wave = tid >> 5;
    const int l    = lane & 15;
    const int half = lane >> 4;
    const int wm   = wave >> 1;            // 0..3  (M direction, 32 rows each)
    const int wn   = wave & 1;             // 0..1  (N direction, 32 cols each)
    const int m0   = blockIdx.x * 128;
    const int n0   = blockIdx.y * 64;

    const float* Abase = A + (size_t)m0 * K;     // + row*K + k
    const float* Bbase = B + n0;                 // + k*Nc + col

    const uint32_t a0B = (uint32_t)(uintptr_t)sA0;   // flat LDS ptr: low 32 bits = LDS offset
    const uint32_t b0B = (uint32_t)(uintptr_t)sB0;
    const uint32_t a1B = (uint32_t)(uintptr_t)sA1;
    const uint32_t b1B = (uint32_t)(uintptr_t)sB1;

    // stage one K-chunk: 4 async b128 for A (128x32) + 2 for B (32x64) per thread
    auto stage = [&](int k0, uint32_t aB, uint32_t bB) {
        #pragma unroll
        for (int r = 0; r < 4; ++r) {
            const int idx = tid + r * 256;
            const int row = idx >> 3;              // 0..127
            const int c4  = (idx & 7) * 4;         // 0..28
            async_b128(aB + (uint32_t)(row * LDA_S + c4) * 4u,
                       Abase + (size_t)row * K + k0 + c4);
        }
        #pragma unroll
        for (int r = 0; r < 2; ++r) {
            const int idx = tid + r * 256;
            const int row = idx >> 4;              // 0..31
            const int c4  = (idx & 15) * 4;        // 0..60
            async_b128(bB + (uint32_t)(row * LDB_S + c4) * 4u,
                       Bbase + (size_t)(k0 + row) * Nc + c4);
        }
    };

    v8f c00 = {}, c01 = {}, c10 = {}, c11 = {};

    stage(0, a0B, b0B);                            // prologue -> buffer 0
    const int nk = K / KC;
    for (int t = 0; t < nk; ++t) {
        if (t + 1 < nk) {
            // prefetch next chunk into the other buffer, then wait for current
            if (t & 1) stage((t + 1) * KC, a0B, b0B);
            else       stage((t + 1) * KC, a1B, b1B);
            asm volatile("s_wait_asynccnt 0x6" ::: "memory");  // in-order: prev 6 done
        } else {
            asm volatile("s_wait_asynccnt 0x0" ::: "memory");
        }
        __syncthreads();

        const float* Af = (t & 1) ? sA1 : sA0;
        const float* Bf = (t & 1) ? sB1 : sB0;
        const int mA0 = (wm * 32 + l) * LDA_S;
        const int nB0 = wn * 32 + l;

        #pragma unroll
        for (int kk = 0; kk < KC; kk += 4) {
            const int ko = kk + 2 * half;
            v2f a0 = *(const v2f*)&Af[mA0 + ko];
            v2f a1 = *(const v2f*)&Af[mA0 + 16 * LDA_S + ko];
            v2f b0 = { Bf[ ko      * LDB_S + nB0],      Bf[(ko + 1) * LDB_S + nB0]      };
            v2f b1 = { Bf[ ko      * LDB_S + nB0 + 16], Bf[(ko + 1) * LDB_S + nB0 + 16] };
            c00 = wmma4(a0, b0, c00);
            c01 = wmma4(a0, b1, c01);
            c10 = wmma4(a1, b0, c10);
            c11 = wmma4(a1, b1, c11);
        }
        __syncthreads();
    }

    const int col = n0 + wn * 32 + l;
    #pragma unroll
    for (int i = 0; i < 8; ++i) {
        const int r = m0 + wm * 32 + i + 8 * half;
        C[(size_t)r        * Nc + col]      = c00[i];
        C[(size_t)r        * Nc + col + 16] = c01[i];
        C[(size_t)(r + 16) * Nc + col]      = c10[i];
        C[(size_t)(r + 16) * Nc + col + 16] = c11[i];
    }
}

// ---------------------------------------------------------------------------
// Kernel 2: RoPE + relayout.  proj[(b*N+n)][colBase + h*64 + d] (ld = row stride)
//           -> out[((b*nheads + h)*N + n)*64 + d]   with rope on (even,odd) pairs
// ---------------------------------------------------------------------------
__global__ void rope_relayout(const float* __restrict__ proj, int ld, int colBase,
                              int nheads, float* __restrict__ out,
                              const float* __restrict__ cosT, const float* __restrict__ sinT,
                              int total)
{
    const int idx = blockIdx.x * blockDim.x + threadIdx.x;
    if (idx >= total) return;
    const int i = idx & 31;                           // pair index 0..31
    const int h = (idx >> 5) % nheads;
    const int n = ((idx >> 5) / nheads) % Nn;
    const int b = idx / (32 * nheads * Nn);

    const float* row = proj + (size_t)(b * Nn + n) * ld + colBase + h * HD;
    const float xe = row[2 * i];
    const float xo = row[2 * i + 1];
    const float cv = cosT[n * 32 + i];
    const float sv = sinT[n * 32 + i];

    float* orow = out + ((size_t)(b * nheads + h) * Nn + n) * HD;
    orow[2 * i]     = xe * cv - xo * sv;
    orow[2 * i + 1] = xe * sv + xo * cv;
}

__global__ void v_relayout(const float* __restrict__ proj, float* __restrict__ out, int total)
{
    const int idx = blockIdx.x * blockDim.x + threadIdx.x;
    if (idx >= total) return;
    const int d = idx & 63;
    const int h = (idx >> 6) % KVH;
    const int n = ((idx >> 6) / KVH) % Nn;
    const int b = idx / (64 * KVH * Nn);
    out[((size_t)(b * KVH + h) * Nn + n) * HD + d] =
        proj[(size_t)(b * Nn + n) * (2 * KVH * HD) + KVH * HD + h * HD + d];
}

// ---------------------------------------------------------------------------
// Kernel 3: flash attention, fp32 WMMA.
// Q [B,H,N,64], K [B,KVH,N,64], V [B,KVH,N,64] -> Out [B*N, H*64]
// grid.x = B*H, grid.y = 16; block = 256 (8 waves); 1 wave = one 16-query tile.
// Reference uses jnp.tile -> Q head h pairs with KV head (h % KVH).
// ---------------------------------------------------------------------------
__global__ __launch_bounds__(256) void attn_flash_wmma(
    const float* __restrict__ Q, const float* __restrict__ Kt,
    const float* __restrict__ V, float* __restrict__ Out)
{
    __shared__ float Plds[8][16 * 18];      // per-wave P bounce buffer (stride 18: 8B-aligned pairs)

    const int tid  = threadIdx.x;
    const int wave = tid >> 5;
    const int lane = tid & 31;
    const int l    = lane & 15;
    const int half = lane >> 4;

    const int bh  = blockIdx.x;             // 0..63
    const int b   = bh / Hh;
    const int h   = bh % Hh;
    const int kvh = h & (KVH - 1);          // jnp.tile semantics: h % 8
    const int t   = blockIdx.y * 8 + wave;  // query tile 0..127
    const int q0  = t * 16;

    const float* Qp = Q  + (size_t)(b * Hh  + h)   * Nn * HD;
    const float* Kp = Kt + (size_t)(b * KVH + kvh) * Nn * HD;
    const float* Vp = V  + (size_t)(b * KVH + kvh) * Nn * HD;
    float* Pw = &Plds[wave][0];

    // Preload all A-fragments of the 16x64 Q tile (16 k-steps of K=4)
    v2f qa[16];
    #pragma unroll
    for (int kx = 0; kx < 16; ++kx)
        qa[kx] = *(const v2f*)&Qp[(size_t)(q0 + l) * HD + kx * 4 + 2 * half];

    v8f o0 = {}, o1 = {}, o2 = {}, o3 = {};
    float mrow[8], lrow[8];
    #pragma unroll
    for (int i = 0; i < 8; ++i) { mrow[i] = -1e30f; lrow[i] = 0.0f; }

    for (int j = 0; j <= t; ++j) {
        const int kv0 = j * 16;

        // S(16q x 16k) = Q * K^T : B-frag b.x = K[kv0+l][k], b.y = K[kv0+l][k+1]
        v8f s = {};
        #pragma unroll
        for (int kx = 0; kx < 16; ++kx) {
            v2f kb = *(const v2f*)&Kp[(size_t)(kv0 + l) * HD + kx * 4 + 2 * half];
            s = wmma4(qa[kx], kb, s);
        }

        // scale + causal mask (only the diagonal tile can cross the diagonal)
        #pragma unroll
        for (int i = 0; i < 8; ++i) {
            float sv = s[i] * 0.125f;                       // HEAD_DIM^-0.5
            const int qrow = q0 + i + 8 * half;
            const int kcol = kv0 + l;
            if (j == t && kcol > qrow) sv = -1e9f;          // NEG_INF
            s[i] = sv;
        }

        // online softmax: row stats per (vgpr i, half); reduce across 16-lane group
        float pscale[8];
        #pragma unroll
        for (int i = 0; i < 8; ++i) {
            float tm = s[i];
            tm = fmaxf(tm, __shfl_xor(tm, 1));
            tm = fmaxf(tm, __shfl_xor(tm, 2));
            tm = fmaxf(tm, __shfl_xor(tm, 4));
            tm = fmaxf(tm, __shfl_xor(tm, 8));
            const float mn = fmaxf(mrow[i], tm);
            const float p  = __expf(s[i] - mn);
            float ts = p;
            ts += __shfl_xor(ts, 1);
            ts += __shfl_xor(ts, 2);
            ts += __shfl_xor(ts, 4);
            ts += __shfl_xor(ts, 8);
            const float sc = __expf(mrow[i] - mn);
            lrow[i]   = lrow[i] * sc + ts;
            mrow[i]   = mn;
            pscale[i] = sc;
            s[i]      = p;
        }
        #pragma unroll
        for (int i = 0; i < 8; ++i) {
            o0[i] *= pscale[i]; o1[i] *= pscale[i];
            o2[i] *= pscale[i]; o3[i] *= pscale[i];
        }

        // D-layout -> A-layout for P via per-wave LDS (DS ops are in-order per wave)
        #pragma unroll
        for (int i = 0; i < 8; ++i)
            Pw[(i + 8 * half) * 18 + l] = s[i];
        __asm__ volatile("s_wait_dscnt 0" ::: "memory");

        // O(16q x 64d) += P(16x16) * V(16k x 64d): 4 col-blocks x 4 k-steps
        #pragma unroll
        for (int kx = 0; kx < 4; ++kx) {
            v2f pa = *(const v2f*)&Pw[l * 18 + kx * 4 + 2 * half];
            const int kr = kv0 + kx * 4 + 2 * half;
            v2f vb0 = { Vp[(size_t)kr * HD +  0 + l], Vp[(size_t)(kr + 1) * HD +  0 + l] };
            v2f vb1 = { Vp[(size_t)kr * HD + 16 + l], Vp[(size_t)(kr + 1) * HD + 16 + l] };
            v2f vb2 = { Vp[(size_t)kr * HD + 32 + l], Vp[(size_t)(kr + 1) * HD + 32 + l] };
            v2f vb3 = { Vp[(size_t)kr * HD + 48 + l], Vp[(size_t)(kr + 1) * HD + 48 + l] };
            o0 = wmma4(pa, vb0, o0);
            o1 = wmma4(pa, vb1, o1);
            o2 = wmma4(pa, vb2, o2);
            o3 = wmma4(pa, vb3, o3);
        }
        __asm__ volatile("s_wait_dscnt 0" ::: "memory");
    }

    // normalize and write Out[(b*N + q)][h*64 + d]
    #pragma unroll
    for (int i = 0; i < 8; ++i) {
        const float inv = 1.0f / lrow[i];
        const int q = q0 + i + 8 * half;
        float* orow = Out + (size_t)(b * Nn + q) * DIM + h * HD;
        orow[ 0 + l] = o0[i] * inv;
        orow[16 + l] = o1[i] * inv;
        orow[32 + l] = o2[i] * inv;
        orow[48 + l] = o3[i] * inv;
    }
}

// ---------------------------------------------------------------------------
// Launcher
// ---------------------------------------------------------------------------
extern "C" void kernel_launch(void* const* d_in, const int* in_sizes, int n_in,
                              void* d_out, int out_size, void* d_ws, size_t ws_size,
                              hipStream_t stream) {
    (void)in_sizes; (void)n_in; (void)out_size; (void)ws_size;

    const float* x   = (const float*)d_in[0];
    const float* Wq  = (const float*)d_in[1];
    const float* Wkv = (const float*)d_in[2];
    const float* Wo  = (const float*)d_in[3];
    const float* rc  = (const float*)d_in[4];
    const float* rs  = (const float*)d_in[5];
    // d_in[6] = mask (recomputed inline), d_in[7] = start_pos (== 0)

    float* ws     = (float*)d_ws;
    float* Qproj  = ws;                                 // 8M floats; reused as attn_out
    float* KVproj = Qproj  + (size_t)ROWS * DIM;        // 4M floats
    float* Qr     = KVproj + (size_t)ROWS * (2 * KVH * HD);
    float* Kr     = Qr     + (size_t)Bc * Hh  * Nn * HD;
    float* Vr     = Kr     + (size_t)Bc * KVH * Nn * HD;

    const dim3 blk(256);

    // QKV projections (fp32 WMMA GEMM, async double-buffered staging)
    gemm_f32_wmma<<<dim3(ROWS / 128, DIM / 64), blk, 0, stream>>>(
        x, Wq, Qproj, ROWS, DIM, DIM);
    gemm_f32_wmma<<<dim3(ROWS / 128, (2 * KVH * HD) / 64), blk, 0, stream>>>(
        x, Wkv, KVproj, ROWS, 2 * KVH * HD, DIM);

    // RoPE + relayout
    const int tq = Bc * Nn * Hh * 32;
    rope_relayout<<<(tq + 255) / 256, blk, 0, stream>>>(Qproj, DIM, 0, Hh, Qr, rc, rs, tq);
    const int tk = Bc * Nn * KVH * 32;
    rope_relayout<<<(tk + 255) / 256, blk, 0, stream>>>(KVproj, 2 * KVH * HD, 0, KVH, Kr, rc, rs, tk);
    const int tv = Bc * Nn * KVH * HD;
    v_relayout<<<(tv + 255) / 256, blk, 0, stream>>>(KVproj, Vr, tv);

    // Flash attention -> reuse Qproj as attn output [B*N, H*64]
    attn_flash_wmma<<<dim3(Bc * Hh, Nn / 16 / 8), blk, 0, stream>>>(Qr, Kr, Vr, Qproj);

    // Output projection
    gemm_f32_wmma<<<dim3(ROWS / 128, DIM / 64), blk, 0, stream>>>(
        Qproj, Wo, (float*)d_out, ROWS, DIM, DIM);
}